// SliceBlock_48215302865260
// MI455X (gfx1250) — compile-verified
//
#include <hip/hip_runtime.h>

typedef __attribute__((ext_vector_type(2))) float v2f;
typedef __attribute__((ext_vector_type(8))) float v8f;

#define BB 4
#define NN 16384
#define EE 65536
#define CC 256
#define MM 64

#define FST 268   // fx LDS row stride (floats): 2*FST%64=24, 8*FST%64=32 -> disjoint half-wave bank sets
#define TST 268   // token LDS row stride
#define SST 72    // sw/ssum/esw LDS row stride

// workspace layout (float offsets)
#define WS_TOK   0                          // B*M*C = 65536
#define WS_NORM  (BB*MM*CC)                 // B*M = 256
#define WS_ESUM  (WS_NORM + BB*MM)          // 256
#define WS_EDOT  (WS_ESUM + BB*MM)          // 256
#define WS_WFP   (WS_EDOT + BB*MM)          // 65536: W_fx packed into B-fragment pairs
#define WS_WXSP  (WS_WFP + CC*CC)           // 16384: (W_x@W_slice) packed pairs
#define WS_BF    (WS_WXSP + CC*MM)          // 64:    (b_x+ph_x)@W_slice + b_slice
#define WS_WEP   (WS_BF + MM)               // 4096:  W_eslice/temp_edge packed pairs
#define WS_ZERO_COUNT WS_WFP                // zero only the accumulator region

__device__ __forceinline__ v8f wmma4(v2f a, v2f b, v8f c) {
  // D = A(16x4 f32) * B(4x16 f32) + C(16x16 f32)
  return __builtin_amdgcn_wmma_f32_16x16x4_f32(false, a, false, b, (short)0, c,
                                               false, false);
}

__device__ __forceinline__ void softmax_rows16(float* buf, int lane) {
  // 16 rows x 64 cols, 2 lanes per row
  const int r = lane >> 1;
  const int cb = (lane & 1) * 32;
  float mx = -3.4e38f;
#pragma unroll 1
  for (int j = 0; j < 32; ++j) mx = fmaxf(mx, buf[r * SST + cb + j]);
  mx = fmaxf(mx, __shfl_xor(mx, 1, 32));
  float sum = 0.f;
#pragma unroll 1
  for (int j = 0; j < 32; ++j) {
    float e = __expf(buf[r * SST + cb + j] - mx);
    buf[r * SST + cb + j] = e;
    sum += e;
  }
  sum += __shfl_xor(sum, 1, 32);
  float inv = 1.0f / sum;
#pragma unroll 1
  for (int j = 0; j < 32; ++j) buf[r * SST + cb + j] *= inv;
}

__global__ void k_init(float* __restrict__ ws) {
  int i = blockIdx.x * blockDim.x + threadIdx.x;
  if (i < WS_ZERO_COUNT) ws[i] = 0.0f;
}

// Pack W_fx and W_eslice into fragment-pair order:
// pair index p = krow*COLS + c, krow = 2*(k/4) + ((k%4)>>1); element comp = k&1.
// Each lane's B-fragment {W[k][c], W[k+1][c]} becomes one aligned 8B load.
__global__ void k_pack(const float* __restrict__ W_fx,
                       const float* __restrict__ W_eslice,
                       const float* __restrict__ temp_edge,
                       float* __restrict__ ws) {
  int t = blockIdx.x * blockDim.x + threadIdx.x;
  if (t < CC * CC) {
    int pair = t >> 1, comp = t & 1;
    int kpair = pair >> 8, c = pair & (CC - 1);
    int k = 4 * (kpair >> 1) + 2 * (kpair & 1) + comp;
    ws[WS_WFP + t] = W_fx[k * CC + c];
  } else if (t < CC * CC + MM * MM) {
    int u = t - CC * CC;
    int pair = u >> 1, comp = u & 1;
    int kpair = pair >> 6, m = pair & (MM - 1);
    int k = 4 * (kpair >> 1) + 2 * (kpair & 1) + comp;
    float tv = temp_edge[k];
    tv = fminf(fmaxf(tv, 0.01f), 5.0f);
    ws[WS_WEP + u] = W_eslice[k * MM + m] / tv;   // fold 1/clip(temp_edge) into B
  }
}

// W_xs[k][m] = sum_j W_x[k][j]*W_slice[j][m] written in packed-pair order;
// bfused[m] = (b_x+ph_x)@W_slice + b_slice
__global__ void k_fuse(const float* __restrict__ W_x, const float* __restrict__ W_slice,
                       const float* __restrict__ b_x, const float* __restrict__ ph_x,
                       const float* __restrict__ b_slice, float* __restrict__ ws) {
  int t = blockIdx.x * blockDim.x + threadIdx.x;
  if (t < CC * MM) {
    int k = t >> 6, m = t & (MM - 1);
    float s = 0.f;
#pragma unroll 4
    for (int j = 0; j < CC; ++j) s += W_x[k * CC + j] * W_slice[j * MM + m];
    int krow = 2 * (k >> 2) + ((k & 3) >> 1);
    ws[WS_WXSP + (krow * MM + m) * 2 + (k & 1)] = s;
  } else if (t < CC * MM + MM) {
    int m = t - CC * MM;
    float s = b_slice[m];
#pragma unroll 4
    for (int j = 0; j < CC; ++j) s += (b_x[j] + ph_x[j]) * W_slice[j * MM + m];
    ws[WS_BF + m] = s;
  }
}

__global__ void __launch_bounds__(256) k_node(
    const float* __restrict__ x, const float* __restrict__ b_fx,
    const float* __restrict__ temp_x, float* __restrict__ ws,
    float* __restrict__ sw_out) {
  extern __shared__ float lds[];
  float* tokL   = lds;                             // MM*TST
  float* fxBase = tokL + MM * TST;                 // 8 waves * 16*FST
  float* swBase = fxBase + 8 * 16 * FST;           // 8 waves * 16*SST

  const int tid  = threadIdx.x;
  const int wv   = tid >> 5;
  const int lane = tid & 31;
  const int half = lane >> 4;
  const int l16  = lane & 15;
  const int b    = blockIdx.y;

  float* fxl = fxBase + wv * 16 * FST;
  float* swl = swBase + wv * 16 * SST;

  for (int i = tid; i < MM * TST; i += 256) tokL[i] = 0.0f;
  __syncthreads();

  // per-column logits temperature (temp_x is [1,1,M])
  float invT[4];
#pragma unroll
  for (int mt = 0; mt < 4; ++mt) {
    float tv = temp_x[mt * 16 + l16];
    tv = fminf(fmaxf(tv, 0.01f), 5.0f);
    invT[mt] = 1.0f / tv;
  }

  const float2* Wfp2 = reinterpret_cast<const float2*>(ws + WS_WFP);
  const float2* Wxp2 = reinterpret_cast<const float2*>(ws + WS_WXSP);
  const float*  bfz  = ws + WS_BF;

#pragma unroll 1
  for (int t = 0; t < 2; ++t) {
    const int tile = blockIdx.x * 16 + t * 8 + wv;   // 0..1023 per batch
    const int n0 = tile * 16;
    const float* xrow = x + ((size_t)b * NN + n0) * CC;

    v8f facc[16] = {};
    v8f lacc[4]  = {};

#pragma unroll 1
    for (int kq = 0; kq < CC / 4; ++kq) {
      const int krow = kq * 2 + half;
      v2f a;
      {
        const float2 av = *(const float2*)(xrow + (size_t)l16 * CC + kq * 4 + 2 * half);
        a.x = av.x; a.y = av.y;
      }
      // load phase: all 20 B-fragments as single b64 loads (one clause, one wait)
      float2 bf[16], bx[4];
#pragma unroll
      for (int ct = 0; ct < 16; ++ct)
        bf[ct] = Wfp2[(size_t)krow * CC + ct * 16 + l16];
#pragma unroll
      for (int mt = 0; mt < 4; ++mt)
        bx[mt] = Wxp2[krow * MM + mt * 16 + l16];
      // compute phase: 20 back-to-back WMMAs
#pragma unroll
      for (int ct = 0; ct < 16; ++ct) {
        v2f bb; bb.x = bf[ct].x; bb.y = bf[ct].y;
        facc[ct] = wmma4(a, bb, facc[ct]);
      }
#pragma unroll
      for (int mt = 0; mt < 4; ++mt) {
        v2f bb; bb.x = bx[mt].x; bb.y = bx[mt].y;
        lacc[mt] = wmma4(a, bb, lacc[mt]);
      }
    }

    // fx_mid = acc + b_fx  -> LDS
#pragma unroll
    for (int ct = 0; ct < 16; ++ct) {
      float bfv = b_fx[ct * 16 + l16];
#pragma unroll
      for (int r = 0; r < 8; ++r)
        fxl[(r + 8 * half) * FST + ct * 16 + l16] = facc[ct][r] + bfv;
    }
    // logits = (acc + bfused) / temp -> LDS
#pragma unroll
    for (int mt = 0; mt < 4; ++mt) {
      float bz = bfz[mt * 16 + l16];
#pragma unroll
      for (int r = 0; r < 8; ++r)
        swl[(r + 8 * half) * SST + mt * 16 + l16] = (lacc[mt][r] + bz) * invT[mt];
    }

    softmax_rows16(swl, lane);

    // store slice_weight to d_out [B,N,M]
    {
      const int r = lane >> 1;
      const int cb = (lane & 1) * 32;
      float* op = sw_out + ((size_t)b * NN + n0 + r) * MM + cb;
#pragma unroll
      for (int j = 0; j < 32; j += 4) {
        float4 v;
        v.x = swl[r * SST + cb + j];     v.y = swl[r * SST + cb + j + 1];
        v.z = swl[r * SST + cb + j + 2]; v.w = swl[r * SST + cb + j + 3];
        *(float4*)(op + j) = v;
      }
    }

    // node_norm partials
#pragma unroll
    for (int h = 0; h < 2; ++h) {
      int m = lane + h * 32;
      float s = 0.f;
#pragma unroll 1
      for (int r2 = 0; r2 < 16; ++r2) s += swl[r2 * SST + m];
      atomicAdd(&ws[WS_NORM + b * MM + m], s);
    }

    // token partial: sw^T(64x16) @ fx(16x256) -> LDS float atomics
#pragma unroll 1
    for (int mt = 0; mt < 4; ++mt) {
      v2f aa[4];
#pragma unroll
      for (int k = 0; k < 4; ++k) {
        const int ka = 4 * k + 2 * half;
        aa[k].x = swl[ka * SST + mt * 16 + l16];
        aa[k].y = swl[(ka + 1) * SST + mt * 16 + l16];
      }
#pragma unroll 1
      for (int ct = 0; ct < 16; ++ct) {
        v2f bb[4];
#pragma unroll
        for (int k = 0; k < 4; ++k) {
          const int ka = 4 * k + 2 * half;
          bb[k].x = fxl[ka * FST + ct * 16 + l16];
          bb[k].y = fxl[(ka + 1) * FST + ct * 16 + l16];
        }
        v8f tacc = {};
#pragma unroll
        for (int k = 0; k < 4; ++k) tacc = wmma4(aa[k], bb[k], tacc);
#pragma unroll
        for (int r = 0; r < 8; ++r)
          atomicAdd(&tokL[(mt * 16 + r + 8 * half) * TST + ct * 16 + l16], tacc[r]);
      }
    }
  }

  __syncthreads();
  for (int i = tid; i < MM * CC; i += 256) {
    int m = i >> 8, c = i & (CC - 1);
    atomicAdd(&ws[WS_TOK + (b * MM + m) * CC + c], tokL[m * TST + c]);
  }
}

__global__ void __launch_bounds__(256) k_edge(
    const float* __restrict__ sw_out, const float* __restrict__ edge_attr,
    const int* __restrict__ edge_idx, const float* __restrict__ b_eslice,
    float* __restrict__ ws) {
  extern __shared__ float lds[];
  const int tid  = threadIdx.x;
  const int wv   = tid >> 5;
  const int lane = tid & 31;
  const int half = lane >> 4;
  const int l16  = lane & 15;
  const int b    = blockIdx.y;

  float* wbase = lds + wv * 2368;
  float* ssum  = wbase;                     // 16*SST
  float* eswl  = wbase + 16 * SST;          // 16*SST
  float* eal   = wbase + 2 * 16 * SST;      // 16
  int*   idxl  = (int*)(eal + 16);          // 32

  const float2* Wep2 = reinterpret_cast<const float2*>(ws + WS_WEP);

  float accS[2] = {0.f, 0.f};
  float accD[2] = {0.f, 0.f};

#pragma unroll 1
  for (int t = 0; t < 8; ++t) {
    const int tile = blockIdx.x * 64 + t * 8 + wv;   // 0..4095 per batch
    const int e0 = tile * 16;

    idxl[lane] = edge_idx[((size_t)b * EE + e0 + (lane >> 1)) * 2 + (lane & 1)];
    if (lane < 16) eal[lane] = edge_attr[(size_t)b * EE + e0 + lane];

    // gather: sw_sum = sw[idx0] + sw[idx1]
#pragma unroll 1
    for (int i = lane; i < 16 * MM; i += 32) {
      int r = i >> 6, m = i & 63;
      int i0 = idxl[2 * r], i1 = idxl[2 * r + 1];
      ssum[r * SST + m] = sw_out[((size_t)b * NN + i0) * MM + m] +
                          sw_out[((size_t)b * NN + i1) * MM + m];
    }

    // e_logits = ssum @ (W_eslice/temp_e)  (temperature pre-folded into Wep)
    v8f eacc[4] = {};
#pragma unroll 1
    for (int kq = 0; kq < MM / 4; ++kq) {
      const int krow = kq * 2 + half;
      v2f a;
      {
        const float2 sv = *(const float2*)(ssum + l16 * SST + kq * 4 + 2 * half);
        a.x = sv.x; a.y = sv.y;
      }
      float2 bw[4];
#pragma unroll
      for (int mt = 0; mt < 4; ++mt) bw[mt] = Wep2[krow * MM + mt * 16 + l16];
#pragma unroll
      for (int mt = 0; mt < 4; ++mt) {
        v2f bb; bb.x = bw[mt].x; bb.y = bw[mt].y;
        eacc[mt] = wmma4(a, bb, eacc[mt]);
      }
    }
#pragma unroll
    for (int mt = 0; mt < 4; ++mt) {
      float bz = b_eslice[mt * 16 + l16];
#pragma unroll
      for (int r = 0; r < 8; ++r)
        eswl[(r + 8 * half) * SST + mt * 16 + l16] = eacc[mt][r] + bz;
    }

    softmax_rows16(eswl, lane);

    // rank-1 reductions: sum(esw), sum(esw * edge_attr)
#pragma unroll
    for (int h = 0; h < 2; ++h) {
      int m = lane + h * 32;
      float s = 0.f, d = 0.f;
#pragma unroll 1
      for (int r2 = 0; r2 < 16; ++r2) {
        float w = eswl[r2 * SST + m];
        s += w;
        d += w * eal[r2];
      }
      accS[h] += s;
      accD[h] += d;
    }
  }

#pragma unroll
  for (int h = 0; h < 2; ++h) {
    int m = lane + h * 32;
    atomicAdd(&ws[WS_ESUM + b * MM + m], accS[h]);
    atomicAdd(&ws[WS_EDOT + b * MM + m], accD[h]);
  }
}

__global__ void k_combine(const float* __restrict__ ws,
                          const float* __restrict__ W_fedge,
                          const float* __restrict__ b_fedge,
                          const float* __restrict__ ph_edge,
                          float* __restrict__ out_tok) {
  int i = blockIdx.x * blockDim.x + threadIdx.x;
  if (i >= BB * MM * CC) return;
  int c = i & (CC - 1);
  int bm = i >> 8;  // b*M + m
  const float scale = (float)NN / (float)EE;
  float tok = ws[WS_TOK + i];
  float nn  = ws[WS_NORM + bm];
  float es  = ws[WS_ESUM + bm];
  float ed  = ws[WS_EDOT + bm];
  float tv  = tok + scale * (ed * W_fedge[c] + es * (b_fedge[c] + ph_edge[c]));
  float nrm = nn + scale * es + 1e-5f;
  out_tok[i] = tv / nrm;
}

extern "C" void kernel_launch(void* const* d_in, const int* in_sizes, int n_in,
                              void* d_out, int out_size, void* d_ws, size_t ws_size,
                              hipStream_t stream) {
  (void)in_sizes; (void)n_in; (void)out_size; (void)ws_size;
  const float* x         = (const float*)d_in[0];
  const float* edge_attr = (const float*)d_in[1];
  const int*   edge_idx  = (const int*)d_in[2];
  const float* W_fx      = (const float*)d_in[3];
  const float* b_fx      = (const float*)d_in[4];
  const float* W_x       = (const float*)d_in[5];
  const float* b_x       = (const float*)d_in[6];
  const float* W_slice   = (const float*)d_in[7];
  const float* b_slice   = (const float*)d_in[8];
  const float* temp_x    = (const float*)d_in[9];
  const float* ph_x      = (const float*)d_in[10];
  const float* W_fedge   = (const float*)d_in[11];
  const float* b_fedge   = (const float*)d_in[12];
  const float* temp_edge = (const float*)d_in[13];
  const float* W_eslice  = (const float*)d_in[14];
  const float* b_eslice  = (const float*)d_in[15];
  const float* ph_edge   = (const float*)d_in[16];

  float* ws      = (float*)d_ws;
  float* out_tok = (float*)d_out;                 // [B,M,C]
  float* sw_out  = out_tok + BB * MM * CC;        // [B,N,M]

  k_init<<<(WS_ZERO_COUNT + 255) / 256, 256, 0, stream>>>(ws);
  k_pack<<<(CC * CC + MM * MM + 255) / 256, 256, 0, stream>>>(W_fx, W_eslice,
                                                              temp_edge, ws);
  k_fuse<<<(CC * MM + MM + 255) / 256, 256, 0, stream>>>(W_x, W_slice, b_x, ph_x,
                                                         b_slice, ws);
  size_t nodeLds = (size_t)(MM * TST + 8 * 16 * FST + 8 * 16 * SST) * sizeof(float);
  k_node<<<dim3(64, BB), 256, nodeLds, stream>>>(x, b_fx, temp_x, ws, sw_out);
  size_t edgeLds = (size_t)(8 * 2368) * sizeof(float);
  k_edge<<<dim3(64, BB), 256, edgeLds, stream>>>(sw_out, edge_attr, edge_idx,
                                                 b_eslice, ws);
  k_combine<<<(BB * MM * CC + 255) / 256, 256, 0, stream>>>(ws, W_fedge, b_fedge,
                                                            ph_edge, out_tok);
}